// BfCtcLoss_36464272343371
// MI455X (gfx1250) — compile-verified
//
#include <hip/hip_runtime.h>
#include <math.h>

#define NEGV (-1e30f)

typedef __attribute__((ext_vector_type(2))) float v2f;
typedef __attribute__((ext_vector_type(8))) float v8f;

namespace {
constexpr int T_ = 512;
constexpr int B_ = 32;
constexpr int V_ = 4096;
constexpr int S_ = 64;
constexpr int L_ = 2 * S_ + 1;   // 129 extended-label states
constexpr int DPT = 160;         // 5 wave32 waves cover 129 states
}

#if __has_builtin(__builtin_amdgcn_wmma_f32_16x16x4_f32)
#define HAVE_WMMA_F32X4 1
#endif

#if __has_builtin(__builtin_amdgcn_global_load_async_to_lds_b32) && \
    __has_builtin(__builtin_amdgcn_s_wait_asynccnt)
#define CTC_ASYNC_GATHER 1
#endif

__device__ __forceinline__ float logaddexpf_(float a, float b) {
  float m = fmaxf(a, b);
  float d = fabsf(a - b);
  return m + log1pf(__expf(-d));
}

// Exact cross-wave32 sum of 64 partials (2 per lane) via one
// V_WMMA_F32_16X16X4_F32 against an all-ones B matrix, then one shfl.
// D[m][n] = sum_k A[m][k]; each lane holds 8 row-sums (rows 0-7 or 8-15),
// so (sum of own 8) + (xor-16 partner) = total, broadcast to all lanes.
__device__ __forceinline__ float wave_sum2(float p0, float p1) {
#if defined(HAVE_WMMA_F32X4)
  v2f a; a.x = p0; a.y = p1;
  v2f bb; bb.x = 1.0f; bb.y = 1.0f;
  v8f c = {};
  v8f d = __builtin_amdgcn_wmma_f32_16x16x4_f32(
      /*neg_a=*/false, a, /*neg_b=*/false, bb,
      /*c_mod=*/(short)0, c, /*reuse_a=*/false, /*reuse_b=*/false);
  float s = d[0] + d[1] + d[2] + d[3] + d[4] + d[5] + d[6] + d[7];
  s += __shfl_xor(s, 16, 32);
  return s;
#else
  float s = p0 + p1;
#pragma unroll
  for (int off = 16; off; off >>= 1) s += __shfl_xor(s, off, 32);
  return s;
#endif
}

// One block per (t,b) row: single global pass (values stay in VGPRs),
// lse[row] = max + log(sum exp(x - max)).
__global__ __launch_bounds__(256) void lse_kernel(const float* __restrict__ acts,
                                                  float* __restrict__ lse) {
  const int row = blockIdx.x;           // t*B + b
  const int tid = threadIdx.x;
  const int wid = tid >> 5, lane = tid & 31;
  const float4* p = reinterpret_cast<const float4*>(acts + (size_t)row * V_);

  float4 v0 = p[tid];
  float4 v1 = p[256 + tid];
  float4 v2 = p[512 + tid];
  float4 v3 = p[768 + tid];

  float m = fmaxf(fmaxf(fmaxf(v0.x, v0.y), fmaxf(v0.z, v0.w)),
                  fmaxf(fmaxf(v1.x, v1.y), fmaxf(v1.z, v1.w)));
  m = fmaxf(m, fmaxf(fmaxf(fmaxf(v2.x, v2.y), fmaxf(v2.z, v2.w)),
                     fmaxf(fmaxf(v3.x, v3.y), fmaxf(v3.z, v3.w))));
#pragma unroll
  for (int off = 16; off; off >>= 1) m = fmaxf(m, __shfl_xor(m, off, 32));

  __shared__ float red[8];
  if (lane == 0) red[wid] = m;
  __syncthreads();
  float M = red[0];
#pragma unroll
  for (int i = 1; i < 8; ++i) M = fmaxf(M, red[i]);
  __syncthreads();

  float p0 = __expf(v0.x - M) + __expf(v0.y - M) + __expf(v0.z - M) + __expf(v0.w - M)
           + __expf(v1.x - M) + __expf(v1.y - M) + __expf(v1.z - M) + __expf(v1.w - M);
  float p1 = __expf(v2.x - M) + __expf(v2.y - M) + __expf(v2.z - M) + __expf(v2.w - M)
           + __expf(v3.x - M) + __expf(v3.y - M) + __expf(v3.z - M) + __expf(v3.w - M);

  float ws = wave_sum2(p0, p1);         // WMMA reduce (EXEC all ones here)
  if (lane == 0) red[wid] = ws;
  __syncthreads();
  if (tid == 0) {
    float S = red[0] + red[1] + red[2] + red[3] + red[4] + red[5] + red[6] + red[7];
    lse[row] = M + logf(S);
  }
}

// One block per batch element: sequential alpha recursion over t,
// double-buffered LDS alpha (1 barrier/iter), next-timestep label-prob
// gather pipelined via async global->LDS copies (ASYNCcnt) when available.
__global__ __launch_bounds__(DPT) void ctc_dp_kernel(
    const float* __restrict__ acts, const int* __restrict__ labels,
    const int* __restrict__ input_lengths, const int* __restrict__ target_lengths,
    const float* __restrict__ lse, float* __restrict__ nll) {
  const int b = blockIdx.x;
  const int tid = threadIdx.x;
  const int s = tid;

  __shared__ float lse_s[T_];
  __shared__ int lab_s[S_];
  __shared__ float alpha[2][L_];
  __shared__ float lpbuf[2][DPT];

  for (int t = tid; t < T_; t += DPT) lse_s[t] = lse[(size_t)t * B_ + b];
  if (tid < S_) lab_s[tid] = labels[b * S_ + tid];
  __syncthreads();

  const int len = input_lengths[b];     // in [T/2, T]; uniform per block
  const int tl = target_lengths[b];

  int ext = 0;
  bool skip = false;
  if (s < L_ && (s & 1)) {
    ext = lab_s[(s - 1) >> 1];          // labels >= 1, so ext != 0 for odd s
    skip = (s >= 3) && (ext != lab_s[(s - 3) >> 1]);
  }

  if (s < L_) {
    float a0 = NEGV;
    if (s < 2) a0 = acts[(size_t)b * V_ + ext] - lse_s[0];   // t = 0
    alpha[0][s] = a0;
  }

#if defined(CTC_ASYNC_GATHER)
  typedef __attribute__((address_space(1))) int gint_t;   // AS1 int* (param 1)
  typedef __attribute__((address_space(3))) int lint_t;   // AS3 int* (param 2)
  if (s < L_)
    __builtin_amdgcn_global_load_async_to_lds_b32(
        (gint_t*)(acts + ((size_t)1 * B_ + b) * V_ + ext),
        (lint_t*)&lpbuf[0][s], 0, 0);
#else
  float x_cur = 0.f, x_next = 0.f;
  if (s < L_) x_cur = acts[((size_t)1 * B_ + b) * V_ + ext];
#endif
  __syncthreads();

  int cur = 0;
  for (int t = 1; t < len; ++t) {
    const int tn = (t + 1 < len) ? (t + 1) : t;
#if defined(CTC_ASYNC_GATHER)
    if (s < L_)
      __builtin_amdgcn_global_load_async_to_lds_b32(
          (gint_t*)(acts + ((size_t)tn * B_ + b) * V_ + ext),
          (lint_t*)&lpbuf[cur ^ 1][s], 0, 0);
    __builtin_amdgcn_s_wait_asynccnt(1);   // gather(t) landed; gather(t+1) in flight
#else
    if (s < L_) x_next = acts[((size_t)tn * B_ + b) * V_ + ext];
#endif
    if (s < L_) {
#if defined(CTC_ASYNC_GATHER)
      float x_cur = ((volatile float*)lpbuf[cur])[s];
#endif
      float a0 = alpha[cur][s];
      float a1 = (s >= 1) ? alpha[cur][s - 1] : NEGV;
      float a2 = skip ? alpha[cur][s - 2] : NEGV;
      alpha[cur ^ 1][s] = logaddexpf_(logaddexpf_(a0, a1), a2) + (x_cur - lse_s[t]);
    }
    __syncthreads();
    cur ^= 1;
#if !defined(CTC_ASYNC_GATHER)
    x_cur = x_next;
#endif
  }

  if (tid == 0) {
    const int hi = 2 * tl;
    const int lo = hi > 0 ? hi - 1 : 0;
    float a = logaddexpf_(alpha[cur][hi], alpha[cur][lo]);
    float v = -a;
    if (__builtin_isinf(v) || v > 1e29f) v = 0.f;
    nll[b] = v;
  }
}

// Deterministic serial sum over B=32 (no float atomics) -> d_out[0].
__global__ void ctc_finalize_kernel(const float* __restrict__ nll,
                                    float* __restrict__ out) {
  if (blockIdx.x == 0 && threadIdx.x == 0) {
    float s = 0.f;
    for (int i = 0; i < B_; ++i) s += nll[i];
    out[0] = s;
  }
}

extern "C" void kernel_launch(void* const* d_in, const int* in_sizes, int n_in,
                              void* d_out, int out_size, void* d_ws, size_t ws_size,
                              hipStream_t stream) {
  const float* acts = (const float*)d_in[0];
  const int* labels = (const int*)d_in[1];
  const int* ilen = (const int*)d_in[2];
  const int* tlen = (const int*)d_in[3];

  float* lse = (float*)d_ws;                 // T*B floats = 64 KB
  float* nll = lse + (size_t)T_ * B_;        // B floats

  lse_kernel<<<T_ * B_, 256, 0, stream>>>(acts, lse);
  ctc_dp_kernel<<<B_, DPT, 0, stream>>>(acts, labels, ilen, tlen, lse, nll);
  ctc_finalize_kernel<<<1, 32, 0, stream>>>(nll, (float*)d_out);

  (void)in_sizes; (void)n_in; (void)out_size; (void)ws_size;
}